// hierEncoder_24816321036706
// MI455X (gfx1250) — compile-verified
//
#include <hip/hip_runtime.h>
#include <math.h>

#define T_SEQ 2048
#define E_DIM 1024
#define H2_DIM 128
#define G3E (3 * E_DIM) // 3072

typedef __attribute__((ext_vector_type(2))) float v2f;
typedef __attribute__((ext_vector_type(8))) float v8f;

__device__ __forceinline__ float sigmoidf_(float x) {
    return 1.0f / (1.0f + expf(-x));
}

// ---------------------------------------------------------------------------
// Phase 0: zero the ping-pong hidden-state buffers (h0 == 0 per reference).
// hbuf layout: [seq(2)][parity(2)][E_DIM] floats.
// ---------------------------------------------------------------------------
__global__ void init_h_kernel(float* __restrict__ hbuf) {
    for (int i = threadIdx.x; i < 2 * 2 * E_DIM; i += blockDim.x)
        hbuf[i] = 0.0f;
}

// ---------------------------------------------------------------------------
// Phase 1: Gi[s, t, :] = g1_w_ih @ emb[tok(s,t)] + g1_b_ih  for all (s,t).
// Cast as GEMM:  A[4096 x 1024] (gathered embeddings, rows 0..2047 = x,
// 2048..4095 = y)  x  B[1024 x 3072] (B[k][n] = w_ih[n][k])  ->  Gi[4096x3072].
// One 16x16 output tile per wave via V_WMMA_F32_16X16X4_F32 (fp32 to match
// the fp32 reference). 8 waves / block share the same M-tile (A reuse).
// ---------------------------------------------------------------------------
__global__ void gi_gemm_kernel(const int* __restrict__ x,
                               const int* __restrict__ y,
                               const float* __restrict__ emb,
                               const float* __restrict__ wih,
                               const float* __restrict__ bih,
                               float* __restrict__ Gi) {
    const int wave   = threadIdx.x >> 5;
    const int lane   = threadIdx.x & 31;
    const int tile   = blockIdx.x * 8 + wave;     // 49152 tiles total
    const int tilesN = G3E / 16;                  // 192 (divisible by 8)
    const int tileM  = tile / tilesN;
    const int tileN  = tile % tilesN;

    const int m     = tileM * 16 + (lane & 15);   // A row for this lane
    const int n     = tileN * 16 + (lane & 15);   // B column for this lane
    const int khalf = (lane >> 4) ? 2 : 0;        // lanes 16-31 hold K+2,K+3

    const int tok = (m < T_SEQ) ? x[m] : y[m - T_SEQ];
    const float* __restrict__ arow = emb + (size_t)tok * E_DIM;
    const float* __restrict__ brow = wih + (size_t)n * E_DIM; // B[k][n]=wih[n][k]

    v8f c = {};
    for (int k = 0; k < E_DIM; k += 4) {
        v2f a = *(const v2f*)(arow + k + khalf); // A[m][k+khalf], A[m][k+khalf+1]
        v2f b = *(const v2f*)(brow + k + khalf); // B[k+khalf][n], B[k+khalf+1][n]
        c = __builtin_amdgcn_wmma_f32_16x16x4_f32(
                /*neg_a=*/false, a, /*neg_b=*/false, b,
                /*c_mod=*/(short)0, c, /*reuse_a=*/false, /*reuse_b=*/false);
    }

    // C/D layout: VGPR v -> M = tileM*16 + v (+8 for lanes 16-31), N = lane&15.
    const float bias  = bih[n];
    const int   rbase = tileM * 16 + ((lane >> 4) ? 8 : 0);
#pragma unroll
    for (int v = 0; v < 8; ++v) {
        Gi[(size_t)(rbase + v) * G3E + n] = c[v] + bias;
    }
}

// ---------------------------------------------------------------------------
// Phase 2: one sequential GRU1 timestep. Launched 2048 times (graph nodes).
// grid = 256 blocks x 256 threads: blocks 0..127 -> seq 0, 128..255 -> seq 1.
// Each block computes 8 output elements (one per wave); each wave does the
// three length-1024 dot products (r,z,n rows of w_hh) with coalesced loads
// and a wave32 __shfl_xor reduction, then the elementwise gate update.
// ---------------------------------------------------------------------------
__global__ void gru1_step_kernel(const float* __restrict__ whh,
                                 const float* __restrict__ bhh,
                                 const float* __restrict__ Gi,
                                 float* __restrict__ hbuf,
                                 int t) {
    __shared__ float hs[E_DIM];

    const int seq = blockIdx.x >> 7;        // 0 or 1
    const int i0  = (blockIdx.x & 127) * 8; // first output element of block
    const int par = t & 1;

    const float* __restrict__ hprev = hbuf + ((size_t)seq * 2 + par) * E_DIM;
    float* __restrict__ hnext       = hbuf + ((size_t)seq * 2 + (par ^ 1)) * E_DIM;

    for (int e = threadIdx.x; e < E_DIM; e += blockDim.x)
        hs[e] = hprev[e];
    __syncthreads();

    const int wave = threadIdx.x >> 5;
    const int lane = threadIdx.x & 31;
    const int i    = i0 + wave;

    const float* __restrict__ wr = whh + (size_t)i * E_DIM;
    const float* __restrict__ wz = whh + (size_t)(E_DIM + i) * E_DIM;
    const float* __restrict__ wn = whh + (size_t)(2 * E_DIM + i) * E_DIM;

    float ar = 0.0f, az = 0.0f, an = 0.0f;
    for (int e = lane; e < E_DIM; e += 32) {
        const float h = hs[e];
        ar = fmaf(wr[e], h, ar);
        az = fmaf(wz[e], h, az);
        an = fmaf(wn[e], h, an);
    }
#pragma unroll
    for (int msk = 16; msk >= 1; msk >>= 1) {
        ar += __shfl_xor(ar, msk, 32);
        az += __shfl_xor(az, msk, 32);
        an += __shfl_xor(an, msk, 32);
    }

    if (lane == 0) {
        const float* __restrict__ gi = Gi + ((size_t)seq * T_SEQ + t) * G3E;
        const float r  = sigmoidf_(gi[i] + ar + bhh[i]);
        const float z  = sigmoidf_(gi[E_DIM + i] + az + bhh[E_DIM + i]);
        const float nn = tanhf(gi[2 * E_DIM + i] + r * (an + bhh[2 * E_DIM + i]));
        hnext[i] = (1.0f - z) * nn + z * hs[i];
    }
}

// ---------------------------------------------------------------------------
// Phase 3: GRU2 (two cells) + Linear(128->32) + ReLU + Linear(32->2) + ReLU
// + log_softmax. Tiny (~1 MFLOP): single block of 384 threads.
// Final GRU1 hidden states sit at parity 0 (T=2048 is even).
// ---------------------------------------------------------------------------
__global__ void head_kernel(const float* __restrict__ hbuf,
                            const float* __restrict__ w_ih2,
                            const float* __restrict__ w_hh2,
                            const float* __restrict__ b_ih2,
                            const float* __restrict__ b_hh2,
                            const float* __restrict__ l1w,
                            const float* __restrict__ l1b,
                            const float* __restrict__ l2w,
                            const float* __restrict__ l2b,
                            float* __restrict__ out) {
    __shared__ float h2[H2_DIM];
    __shared__ float gi[3 * H2_DIM];
    __shared__ float gh[3 * H2_DIM];
    __shared__ float o1[32];
    __shared__ float o2[2];

    const int tid = threadIdx.x; // 0..383 == one gate-row each
    if (tid < H2_DIM) h2[tid] = 0.0f;
    __syncthreads();

    for (int s = 0; s < 2; ++s) {
        const float* __restrict__ hx = hbuf + (size_t)s * 2 * E_DIM; // parity 0
        float accI = b_ih2[tid];
        const float* __restrict__ wi = w_ih2 + (size_t)tid * E_DIM;
        for (int e = 0; e < E_DIM; ++e) accI = fmaf(wi[e], hx[e], accI);
        float accH = b_hh2[tid];
        const float* __restrict__ wh = w_hh2 + (size_t)tid * H2_DIM;
        for (int e = 0; e < H2_DIM; ++e) accH = fmaf(wh[e], h2[e], accH);
        gi[tid] = accI;
        gh[tid] = accH;
        __syncthreads();
        if (tid < H2_DIM) {
            const float r  = sigmoidf_(gi[tid] + gh[tid]);
            const float z  = sigmoidf_(gi[H2_DIM + tid] + gh[H2_DIM + tid]);
            const float nn = tanhf(gi[2 * H2_DIM + tid] + r * gh[2 * H2_DIM + tid]);
            h2[tid] = (1.0f - z) * nn + z * h2[tid];
        }
        __syncthreads();
    }

    if (tid < 32) {
        float a = l1b[tid];
        for (int e = 0; e < H2_DIM; ++e) a = fmaf(l1w[tid * H2_DIM + e], h2[e], a);
        o1[tid] = fmaxf(a, 0.0f);
    }
    __syncthreads();
    if (tid < 2) {
        float a = l2b[tid];
        for (int e = 0; e < 32; ++e) a = fmaf(l2w[tid * 32 + e], o1[e], a);
        o2[tid] = fmaxf(a, 0.0f);
    }
    __syncthreads();
    if (tid == 0) {
        const float mx  = fmaxf(o2[0], o2[1]);
        const float lse = mx + logf(expf(o2[0] - mx) + expf(o2[1] - mx));
        out[0] = o2[0] - lse;
        out[1] = o2[1] - lse;
    }
}

// ---------------------------------------------------------------------------
extern "C" void kernel_launch(void* const* d_in, const int* in_sizes, int n_in,
                              void* d_out, int out_size, void* d_ws, size_t ws_size,
                              hipStream_t stream) {
    (void)in_sizes; (void)n_in; (void)out_size; (void)ws_size;

    const int*   x       = (const int*)d_in[0];
    const int*   y       = (const int*)d_in[1];
    const float* emb     = (const float*)d_in[2];
    const float* g1_w_ih = (const float*)d_in[3];
    const float* g1_w_hh = (const float*)d_in[4];
    const float* g1_b_ih = (const float*)d_in[5];
    const float* g1_b_hh = (const float*)d_in[6];
    const float* g2_w_ih = (const float*)d_in[7];
    const float* g2_w_hh = (const float*)d_in[8];
    const float* g2_b_ih = (const float*)d_in[9];
    const float* g2_b_hh = (const float*)d_in[10];
    const float* l1_w    = (const float*)d_in[11];
    const float* l1_b    = (const float*)d_in[12];
    const float* l2_w    = (const float*)d_in[13];
    const float* l2_b    = (const float*)d_in[14];
    float*       out     = (float*)d_out;

    // Workspace layout: Gi[2*T*3E] floats (50.3 MB) then hbuf[2][2][E].
    float* Gi   = (float*)d_ws;
    float* hbuf = Gi + (size_t)2 * T_SEQ * G3E;

    init_h_kernel<<<1, 256, 0, stream>>>(hbuf);

    // 49152 tiles / 8 waves per block = 6144 blocks.
    gi_gemm_kernel<<<6144, 256, 0, stream>>>(x, y, emb, g1_w_ih, g1_b_ih, Gi);

    // Sequential recurrence: 2048 graph-captured launches, both sequences
    // batched per launch (grid.y folded into blockIdx.x).
    for (int t = 0; t < T_SEQ; ++t)
        gru1_step_kernel<<<256, 256, 0, stream>>>(g1_w_hh, g1_b_hh, Gi, hbuf, t);

    head_kernel<<<1, 384, 0, stream>>>(hbuf, g2_w_ih, g2_w_hh, g2_b_ih, g2_b_hh,
                                       l1_w, l1_b, l2_w, l2_b, out);
}